// Model_69767448756490
// MI455X (gfx1250) — compile-verified
//
#include <hip/hip_runtime.h>
#include <stdint.h>

// out[i] = idx[3i]*8192 + idx[3i+1]*16 + idx[3i+2]   (int32)
// Pure HBM-streaming kernel: 192 MB read + 64 MB write ~= 11 us at 23.3 TB/s.
// CDNA5 path: async global->LDS staging (b128, ASYNCcnt double-buffer, NT
// temporal hint) + ds_load_b128 readback + non-temporal b128 stores.

typedef __attribute__((ext_vector_type(4))) int v4i;

#define BLOCK 256
#define TUP_PER_THREAD 4
#define TUP_PER_CHUNK (BLOCK * TUP_PER_THREAD)   // 1024 tuples per chunk
#define INTS_PER_CHUNK (TUP_PER_CHUNK * 3)       // 3072 ints = 12 KB per buffer
#define BYTES_PER_LANE 48                        // 4 tuples * 3 ints * 4 B

__global__ __launch_bounds__(BLOCK)
void linidx_main(const int* __restrict__ idx, int* __restrict__ out, int numChunks) {
    __shared__ __align__(16) int lds[2][INTS_PER_CHUNK];

    const unsigned tid      = threadIdx.x;
    const unsigned laneByte = tid * BYTES_PER_LANE;

    unsigned ldsBase[2];
    ldsBase[0] = (unsigned)(size_t)(&lds[0][0]);   // low 32 bits of flat addr = LDS offset
    ldsBase[1] = (unsigned)(size_t)(&lds[1][0]);

    int c = (int)blockIdx.x;
    if (c >= numChunks) return;

    // Prologue: stage chunk c into buffer 0.
    // INST_OFFSET is added to BOTH the LDS and the global address, so one
    // lds-base VGPR + one lane-offset VGPR covers all three 16B pieces.
    // Read-once stream -> TH_LOAD_NT.
    {
        const int* cp = idx + (size_t)c * INTS_PER_CHUNK;
        asm volatile(
            "global_load_async_to_lds_b128 %0, %1, %2 offset:0 th:TH_LOAD_NT\n\t"
            "global_load_async_to_lds_b128 %0, %1, %2 offset:16 th:TH_LOAD_NT\n\t"
            "global_load_async_to_lds_b128 %0, %1, %2 offset:32 th:TH_LOAD_NT"
            :
            : "v"(ldsBase[0] + laneByte), "v"(laneByte), "s"(cp)
            : "memory");
    }

    int p = 0;
    while (true) {
        const int nc = c + (int)gridDim.x;
        if (nc < numChunks) {
            // s_wait_dscnt 0: the buffer we are about to overwrite was read by
            // ds_load_b128 last iteration; async LDS writes are NOT ordered
            // against outstanding DS reads, so retire them first.
            // Then issue next chunk and wait until only those 3 async loads
            // remain outstanding -> current buffer is complete.
            const int* cp = idx + (size_t)nc * INTS_PER_CHUNK;
            asm volatile(
                "s_wait_dscnt 0x0\n\t"
                "global_load_async_to_lds_b128 %0, %1, %2 offset:0 th:TH_LOAD_NT\n\t"
                "global_load_async_to_lds_b128 %0, %1, %2 offset:16 th:TH_LOAD_NT\n\t"
                "global_load_async_to_lds_b128 %0, %1, %2 offset:32 th:TH_LOAD_NT"
                :
                : "v"(ldsBase[p ^ 1] + laneByte), "v"(laneByte), "s"(cp)
                : "memory");
            asm volatile("s_wait_asynccnt 0x3" ::: "memory");
        } else {
            asm volatile("s_wait_asynccnt 0x0" ::: "memory");
        }

        // Read back this lane's 4 tuples (3 x ds_load_b128) and combine.
        const v4i* my = (const v4i*)(&lds[p][tid * 12]);
        const v4i a = my[0];
        const v4i b = my[1];
        const v4i d = my[2];
        v4i o;
        o.x = a.x * 8192 + a.y * 16 + a.z;
        o.y = a.w * 8192 + b.x * 16 + b.y;
        o.z = b.z * 8192 + b.w * 16 + d.x;
        o.w = d.y * 8192 + d.z * 16 + d.w;

        // Write-once stream: non-temporal b128 store.
        __builtin_nontemporal_store(o, (v4i*)(out + (size_t)c * TUP_PER_CHUNK + tid * 4));

        if (nc >= numChunks) break;
        p ^= 1;
        c = nc;
    }
}

__global__ void linidx_tail(const int* __restrict__ idx, int* __restrict__ out,
                            int start, int total) {
    const int i = start + (int)(blockIdx.x * blockDim.x + threadIdx.x);
    if (i < total) {
        out[i] = idx[3 * i] * 8192 + idx[3 * i + 1] * 16 + idx[3 * i + 2];
    }
}

extern "C" void kernel_launch(void* const* d_in, const int* in_sizes, int n_in,
                              void* d_out, int out_size, void* d_ws, size_t ws_size,
                              hipStream_t stream) {
    (void)in_sizes; (void)n_in; (void)d_ws; (void)ws_size;

    const int* idx = (const int*)d_in[0];   // indices, [N][3] int32
    int* out = (int*)d_out;                 // int32 written through harness buffer
    const int N = out_size;

    const int numChunks = N / TUP_PER_CHUNK;
    if (numChunks > 0) {
        int blocks = numChunks < 4096 ? numChunks : 4096;  // ~4 chunks/block pipeline
        linidx_main<<<blocks, BLOCK, 0, stream>>>(idx, out, numChunks);
    }
    const int done = numChunks * TUP_PER_CHUNK;
    if (done < N) {
        const int rem = N - done;
        linidx_tail<<<(rem + 255) / 256, 256, 0, stream>>>(idx, out, done, rem);
    }
}